// PolyDecoder_8916352107077
// MI455X (gfx1250) — compile-verified
//
#include <hip/hip_runtime.h>
#include <hip/hip_bf16.h>

// PolyDecoder: out = polyfeat(z) @ W + b
//   z:[8192,16] f32, W:[4369,4096] f32, b:[4096] f32, out:[8192,4096] f32
// K-dim 4369 = 1 (const) + 16 (deg1) + 256 (deg2) + 4096 (deg3).
// Each degree segment is a multiple of 4 -> run as its own f32 WMMA GEMM
// (V_WMMA_F32_16X16X4_F32), synthesizing A-fragments from the 16x16 z tile.

typedef __attribute__((ext_vector_type(2))) float v2f;
typedef __attribute__((ext_vector_type(8))) float v8f;

#define TILE_M          16
#define WAVES_PB        8
#define NPW             4                          // 16-col tiles per wave
#define COLS_PER_WAVE   (NPW * 16)                 // 64
#define COLS_PER_BLOCK  (WAVES_PB * COLS_PER_WAVE) // 512
#define ZPITCH          18   // LDS row pitch (even for b64 align, odd/32 for banks)

__global__ __launch_bounds__(WAVES_PB * 32, 1)
void poly_decoder_wmma_f32(const float* __restrict__ z,
                           const float* __restrict__ W,
                           const float* __restrict__ bias,
                           float*       __restrict__ out,
                           int D)
{
    __shared__ float zsh[16 * ZPITCH];

    const int tid  = threadIdx.x;
    const int lane = tid & 31;
    const int wave = tid >> 5;
    const int kh   = (lane >> 4) & 1;   // lane-half selects K = base + 2*kh (+0/+1)
    const int m    = lane & 15;         // row within tile (A) / col within tile (B)

    const int m0      = blockIdx.x * TILE_M;                       // batch rows
    const int colBase = blockIdx.y * COLS_PER_BLOCK + wave * COLS_PER_WAVE + m;

    // Stage the 16x16 z tile into LDS (one element per thread).
    {
        const int r = tid >> 4, c = tid & 15;
        zsh[r * ZPITCH + c] = z[(m0 + r) * 16 + c];
    }
    __syncthreads();

    const float* zrow = &zsh[m * ZPITCH];

    // Per-lane: the 4 low-digit z pairs this lane-half ever needs (registers).
    float2 zp[4];
#pragma unroll
    for (int j4 = 0; j4 < 4; ++j4)
        zp[j4] = *(const float2*)(zrow + j4 * 4 + 2 * kh);

    v8f acc[NPW];
#pragma unroll
    for (int t = 0; t < NPW; ++t)
#pragma unroll
        for (int r = 0; r < 8; ++r) acc[t][r] = 0.0f;

    const float* W1 = W + (size_t)1   * D;   // deg-1 rows, K=16
    const float* W2 = W + (size_t)17  * D;   // deg-2 rows, K=256
    const float* W3 = W + (size_t)273 * D;   // deg-3 rows, K=4096

    // One K=4 step against NPW column tiles. A value k index = grow + 2*kh (+1),
    // matching B rows loaded below (B layout: VGPR r, lane-half h -> K = r + 2h).
    auto kstep = [&](const float* __restrict__ Wseg, int grow, float ax, float ay) {
        v2f a; a.x = ax; a.y = ay;
        const float* wp = Wseg + (size_t)(grow + 2 * kh) * D + colBase;
#pragma unroll
        for (int t = 0; t < NPW; ++t) {
            v2f bf;
            bf.x = wp[t * 16];            // row grow+2kh
            bf.y = wp[t * 16 + D];        // row grow+2kh+1
            acc[t] = __builtin_amdgcn_wmma_f32_16x16x4_f32(
                /*neg_a=*/false, a, /*neg_b=*/false, bf,
                /*c_mod=*/(short)0, acc[t], /*reuse_a=*/false, /*reuse_b=*/false);
        }
    };

    // ---- degree 1: K = 16, feature p -> z[m][p] ----
#pragma unroll
    for (int j4 = 0; j4 < 4; ++j4)
        kstep(W1, j4 * 4, zp[j4].x, zp[j4].y);

    // ---- degree 2: K = 256, feature p = i*16 + l -> z_i * z_l ----
    for (int i = 0; i < 16; ++i) {
        const float zi = zrow[i];
#pragma unroll
        for (int j4 = 0; j4 < 4; ++j4)
            kstep(W2, i * 16 + j4 * 4, zi * zp[j4].x, zi * zp[j4].y);
    }

    // ---- degree 3: K = 4096, feature p = i*256 + j*16 + l -> z_i * z_j * z_l ----
    for (int i = 0; i < 16; ++i) {
        const float zi = zrow[i];
        // Pull the next i-block of W3 toward the caches while this one computes.
        __builtin_prefetch(W3 + (size_t)(((i + 1) & 15) * 256) * D + colBase, 0, 3);
        for (int j = 0; j < 16; ++j) {
            const float h = zi * zrow[j];
            const int base = (i * 16 + j) * 16;
#pragma unroll
            for (int j4 = 0; j4 < 4; ++j4)
                kstep(W3, base + j4 * 4, h * zp[j4].x, h * zp[j4].y);
        }
    }

    // ---- epilogue: + bias + constant-feature row W[0,:], then store ----
    // C/D layout: VGPR r -> row m0 + r + 8*kh, col = colBase (+ t*16).
#pragma unroll
    for (int t = 0; t < NPW; ++t) {
        const int col = colBase + t * 16;
        const float c0 = bias[col] + W[col];      // 1.0 * W[0][col] + b[col]
        float* op = out + (size_t)(m0 + 8 * kh) * D + col;
#pragma unroll
        for (int r = 0; r < 8; ++r)
            op[(size_t)r * D] = acc[t][r] + c0;
    }
}

extern "C" void kernel_launch(void* const* d_in, const int* in_sizes, int n_in,
                              void* d_out, int out_size, void* d_ws, size_t ws_size,
                              hipStream_t stream) {
    const float* z    = (const float*)d_in[0];   // [B,16]
    const float* W    = (const float*)d_in[1];   // [4369, D]
    const float* bias = (const float*)d_in[2];   // [D]
    float* out        = (float*)d_out;

    const int L = 16;
    const int D = in_sizes[2];          // 4096
    const int B = in_sizes[0] / L;      // 8192

    dim3 block(WAVES_PB * 32);
    dim3 grid(B / TILE_M, D / COLS_PER_BLOCK);   // (512, 8)
    hipLaunchKernelGGL(poly_decoder_wmma_f32, grid, block, 0, stream,
                       z, W, bias, out, D);

    (void)n_in; (void)out_size; (void)d_ws; (void)ws_size;
}